// AttentionGoalState_6322191859974
// MI455X (gfx1250) — compile-verified
//
#include <hip/hip_runtime.h>

typedef __attribute__((ext_vector_type(2))) float v2f;
typedef __attribute__((ext_vector_type(8))) float v8f;

#define WMMA_F32(a, b, c) \
    __builtin_amdgcn_wmma_f32_16x16x4_f32(false, (a), false, (b), (short)0, (c), false, false)

// ---------------------------------------------------------------------------
// WMMA f32 16x16x4, 2x2 register-blocked K-accumulation (32x32 C tile / wave).
// A fragment: lane holds A[m, kb], A[m, kb+1] with m = lane&15, kb = 2*(lane>>4)
// B fragment: lane holds B[kb, n], B[kb+1, n] with n = lane&15
// C/D: vgpr r -> row r + 8*(lane>>4), col lane&15
// Per k-step: 4 operand fragments feed 4 WMMAs (2x data reuse vs 1x1).
// ---------------------------------------------------------------------------
__device__ __forceinline__ void wmma_tile2x2(const float* __restrict__ pa0,
                                             const float* __restrict__ pa1, long sAk,
                                             const float* __restrict__ pb0,
                                             const float* __restrict__ pb1, long sBk,
                                             int K, v8f& c00, v8f& c01, v8f& c10, v8f& c11)
{
#pragma unroll 2
    for (int k = 0; k < K; k += 4) {
        v2f a0, a1, b0, b1;
        a0.x = pa0[0]; a0.y = pa0[sAk];
        a1.x = pa1[0]; a1.y = pa1[sAk];
        b0.x = pb0[0]; b0.y = pb0[sBk];
        b1.x = pb1[0]; b1.y = pb1[sBk];
        c00 = WMMA_F32(a0, b0, c00);
        c01 = WMMA_F32(a0, b1, c01);
        c10 = WMMA_F32(a1, b0, c10);
        c11 = WMMA_F32(a1, b1, c11);
        pa0 += 4 * sAk; pa1 += 4 * sAk;
        pb0 += 4 * sBk; pb1 += 4 * sBk;
    }
}

// ---------------------------------------------------------------------------
// Generic batched GEMM: C[b] = A (MxK, row major, shared) * B[b] (KxN)
// One wave per 32x32 tile. M,N multiples of 32.
// ---------------------------------------------------------------------------
__global__ void k_gemm(const float* __restrict__ A, const float* __restrict__ B,
                       float* __restrict__ C, int K, long sBk, long sBbatch,
                       long sCbatch, int ldc)
{
    int lane = threadIdx.x & 31;
    int l15  = lane & 15;
    int kb   = 2 * (lane >> 4);
    int m0   = blockIdx.y * 32 + l15;
    int n0   = blockIdx.x * 32 + l15;
    const float* Bb = B + (long)blockIdx.z * sBbatch;
    v8f c00 = {}, c01 = {}, c10 = {}, c11 = {};
    wmma_tile2x2(A + (long)m0 * K + kb, A + (long)(m0 + 16) * K + kb, 1,
                 Bb + n0 + (long)kb * sBk, Bb + n0 + 16 + (long)kb * sBk, sBk,
                 K, c00, c01, c10, c11);
    float* Cb = C + (long)blockIdx.z * sCbatch;
    int row0 = blockIdx.y * 32 + 8 * (lane >> 4);
#pragma unroll
    for (int r = 0; r < 8; ++r) {
        Cb[(long)(row0 + r) * ldc + n0]           = c00[r];
        Cb[(long)(row0 + r) * ldc + n0 + 16]      = c01[r];
        Cb[(long)(row0 + r + 16) * ldc + n0]      = c10[r];
        Cb[(long)(row0 + r + 16) * ldc + n0 + 16] = c11[r];
    }
}

// ---------------------------------------------------------------------------
// Conv3d 3x3x3 VALID as implicit GEMM, fused bias+ReLU store.
// X: [4,1024,4,16,16]  W: [512,1024,3,3,3]  out: [4,512,392]
// ---------------------------------------------------------------------------
__global__ void k_conv3d(const float* __restrict__ X, const float* __restrict__ W,
                         const float* __restrict__ bias, float* __restrict__ out)
{
    int lane = threadIdx.x & 31;
    int l15  = lane & 15;
    int kb   = 2 * (lane >> 4);
    int b    = blockIdx.z;
    int m0   = blockIdx.y * 32 + l15;
    int n0   = blockIdx.x * 32 + l15;
    int n1   = n0 + 16;
    int nn0  = n0 < 392 ? n0 : 391;     // clamp gather for edge tile
    int nn1  = n1 < 392 ? n1 : 391;
    int t0a = nn0 / 196, ra = nn0 % 196;
    int t0b = nn1 / 196, rb = nn1 % 196;
    long base0 = (long)b * 4194304 + t0a * 256 + (ra / 14) * 16 + (ra % 14);
    long base1 = (long)b * 4194304 + t0b * 256 + (rb / 14) * 16 + (rb % 14);
    v8f c00 = {}, c01 = {}, c10 = {}, c11 = {};
    for (int kd = 0; kd < 3; ++kd)
        for (int kh = 0; kh < 3; ++kh)
            for (int kw = 0; kw < 3; ++kw) {
                int tap  = (kd * 3 + kh) * 3 + kw;
                long dof = kd * 256 + kh * 16 + kw + (long)kb * 1024;
                wmma_tile2x2(W + (long)m0 * 27648 + tap + (long)kb * 27,
                             W + (long)(m0 + 16) * 27648 + tap + (long)kb * 27, 27,
                             X + base0 + dof, X + base1 + dof, 1024,
                             1024, c00, c01, c10, c11);
            }
    int row0 = blockIdx.y * 32 + 8 * (lane >> 4);
#pragma unroll
    for (int r = 0; r < 8; ++r) {
        int oA = row0 + r, oB = row0 + r + 16;
        if (n0 < 392) {
            out[((long)b * 512 + oA) * 392 + n0] = fmaxf(c00[r] + bias[oA], 0.f);
            out[((long)b * 512 + oB) * 392 + n0] = fmaxf(c10[r] + bias[oB], 0.f);
        }
        if (n1 < 392) {
            out[((long)b * 512 + oA) * 392 + n1] = fmaxf(c01[r] + bias[oA], 0.f);
            out[((long)b * 512 + oB) * 392 + n1] = fmaxf(c11[r] + bias[oB], 0.f);
        }
    }
}

// ---------------------------------------------------------------------------
// ConvTranspose2d(k=2,s=2) as GEMM: rows m = o*4 + k*2 + l, cols = pixels.
// Wt: [C, O=512, 2, 2] -> A[m,c] = Wt[c*2048 + m].  Xin: [BT, C, H, W].
// out: [BT, 512, 2H, 2W], +bias, optional ReLU.
// ---------------------------------------------------------------------------
__global__ void k_convt2(const float* __restrict__ Wt, const float* __restrict__ Xin,
                         const float* __restrict__ bias, float* __restrict__ out,
                         int K, int pixPerBT, int Win, int doRelu)
{
    int lane = threadIdx.x & 31;
    int l15  = lane & 15;
    int kb   = 2 * (lane >> 4);
    int m0   = blockIdx.y * 32 + l15;
    int p0   = blockIdx.x * 32 + l15;
    int p1   = p0 + 16;
    int bt0 = p0 / pixPerBT, hw0 = p0 % pixPerBT;
    int bt1 = p1 / pixPerBT, hw1 = p1 % pixPerBT;
    long col0 = (long)bt0 * K * pixPerBT + hw0;
    long col1 = (long)bt1 * K * pixPerBT + hw1;
    v8f c00 = {}, c01 = {}, c10 = {}, c11 = {};
    wmma_tile2x2(Wt + m0 + (long)kb * 2048, Wt + m0 + 16 + (long)kb * 2048, 2048,
                 Xin + col0 + (long)kb * pixPerBT, Xin + col1 + (long)kb * pixPerBT,
                 pixPerBT, K, c00, c01, c10, c11);
    int row0 = blockIdx.y * 32 + 8 * (lane >> 4);
    int h0 = hw0 / Win, w0 = hw0 % Win;
    int h1 = hw1 / Win, w1 = hw1 % Win;
    int Wout = 2 * Win;
    long ob0 = (long)bt0 * 512 * 4 * pixPerBT;
    long ob1 = (long)bt1 * 512 * 4 * pixPerBT;
#pragma unroll
    for (int r = 0; r < 8; ++r) {
#pragma unroll
        for (int half = 0; half < 2; ++half) {
            int mm = row0 + r + 16 * half;
            int o = mm >> 2, kk = (mm >> 1) & 1, ll = mm & 1;
            float bia = bias[o];
            float vA = (half ? c10[r] : c00[r]) + bia;
            float vB = (half ? c11[r] : c01[r]) + bia;
            if (doRelu) { vA = fmaxf(vA, 0.f); vB = fmaxf(vB, 0.f); }
            out[ob0 + (long)o * 4 * pixPerBT + (2 * h0 + kk) * Wout + (2 * w0 + ll)] = vA;
            out[ob1 + (long)o * 4 * pixPerBT + (2 * h1 + kk) * Wout + (2 * w1 + ll)] = vB;
        }
    }
}

// ---------- elementwise / reduction helpers ----------
__global__ void k_transpose(const float* __restrict__ ctx, float* __restrict__ x)
{
    long i = (long)blockIdx.x * blockDim.x + threadIdx.x;
    if (i >= 4L * 1024 * 1024) return;
    int n = (int)(i & 1023), c = (int)((i >> 10) & 1023), b = (int)(i >> 20);
    int t = n >> 8, hw = n & 255;
    x[i] = ctx[(((long)b * 4 + t) * 1024 + c) * 256 + hw];
}

__global__ void k_addrelu(const float* __restrict__ x, const float* __restrict__ o,
                          const float* __restrict__ ob, float* __restrict__ y)
{
    long i = (long)blockIdx.x * blockDim.x + threadIdx.x;
    if (i >= 4L * 1024 * 1024) return;
    int c = (int)((i >> 10) & 1023);
    y[i] = x[i] + fmaxf(o[i] + ob[c], 0.f);
}

// Training-mode BatchNorm3d over (B,T,H,W) per channel, in place. 1 block/channel.
__global__ void k_bn(float* __restrict__ y, const float* __restrict__ g,
                     const float* __restrict__ beta)
{
    __shared__ float sA[256], sB[256];
    int c = blockIdx.x;
    float sum = 0.f, sq = 0.f;
    for (int idx = threadIdx.x; idx < 4096; idx += 256) {
        int b = idx >> 10, n = idx & 1023;
        float v = y[(long)b * 1048576 + c * 1024 + n];
        sum += v; sq += v * v;
    }
    sA[threadIdx.x] = sum; sB[threadIdx.x] = sq;
    __syncthreads();
    for (int s = 128; s > 0; s >>= 1) {
        if (threadIdx.x < s) { sA[threadIdx.x] += sA[threadIdx.x + s];
                               sB[threadIdx.x] += sB[threadIdx.x + s]; }
        __syncthreads();
    }
    float m   = sA[0] * (1.f / 4096.f);
    float var = sB[0] * (1.f / 4096.f) - m * m;
    float scale = rsqrtf(var + 1e-5f) * g[c];
    float shift = beta[c] - m * scale;
    for (int idx = threadIdx.x; idx < 4096; idx += 256) {
        int b = idx >> 10, n = idx & 1023;
        long a = (long)b * 1048576 + c * 1024 + n;
        y[a] = y[a] * scale + shift;
    }
}

__global__ void k_tpred(const float* __restrict__ tp, float* __restrict__ gp)
{
    __shared__ float s[256];
    int bo = blockIdx.x;
    float sum = 0.f;
    for (int n = threadIdx.x; n < 392; n += 256) sum += tp[(long)bo * 392 + n];
    s[threadIdx.x] = sum; __syncthreads();
    for (int st = 128; st > 0; st >>= 1) {
        if (threadIdx.x < st) s[threadIdx.x] += s[threadIdx.x + st];
        __syncthreads();
    }
    if (threadIdx.x == 0) gp[bo] = s[0] * (1.f / 392.f);
}

__global__ void k_pe(float* __restrict__ pe)
{
    int i = blockIdx.x * 256 + threadIdx.x;
    if (i >= 1024 * 512) return;
    int n = i >> 9, c = i & 511;
    float div = expf((float)(c & ~1) * (-9.2103403719761836f / 512.f));
    float ang = (float)n * div;
    pe[i] = (c & 1) ? cosf(ang) : sinf(ang);
}

// SG dot + softmax over 1024 positions + PE-augmented weighted sum. 1 block/bt.
__global__ void k_attn(const float* __restrict__ s2, const float* __restrict__ gp,
                       const float* __restrict__ pe, float* __restrict__ sga)
{
    __shared__ float goal[512];
    __shared__ float sg[1024];
    __shared__ float red[256];
    int bt = blockIdx.x, b = bt / 6, t = threadIdx.x;
    for (int c = t; c < 512; c += 256) goal[c] = gp[b * 512 + c];
    __syncthreads();
    const float* S = s2 + (long)bt * 524288;
    for (int q = 0; q < 4; ++q) {
        int n = t + 256 * q;
        float acc = 0.f;
        for (int c = 0; c < 512; ++c) acc += S[(long)c * 1024 + n] * goal[c];
        sg[n] = acc;
    }
    __syncthreads();
    float mx = fmaxf(fmaxf(sg[t], sg[t + 256]), fmaxf(sg[t + 512], sg[t + 768]));
    red[t] = mx; __syncthreads();
    for (int s = 128; s > 0; s >>= 1) { if (t < s) red[t] = fmaxf(red[t], red[t + s]); __syncthreads(); }
    mx = red[0]; __syncthreads();
    float lsum = 0.f;
    for (int q = 0; q < 4; ++q) {
        int n = t + 256 * q;
        float e = expf(sg[n] - mx); sg[n] = e; lsum += e;
    }
    red[t] = lsum; __syncthreads();
    for (int s = 128; s > 0; s >>= 1) { if (t < s) red[t] += red[t + s]; __syncthreads(); }
    float inv = 1.f / red[0]; __syncthreads();
    for (int q = 0; q < 4; ++q) sg[t + 256 * q] *= inv;
    __syncthreads();
    for (int c = t; c < 512; c += 256) {
        float acc = 0.f;
        for (int n = 0; n < 1024; ++n)
            acc += sg[n] * (S[(long)c * 1024 + n] + pe[n * 512 + c]);
        sga[bt * 512 + c] = acc;
    }
}

// Small FC head: out[r,j] = act(X[r,:] . W[j,:] + b[j]), K=512.
__global__ void k_head(const float* __restrict__ X, const float* __restrict__ W,
                       const float* __restrict__ bias, float* __restrict__ out,
                       int rows, int cols, int doRelu)
{
    int i = blockIdx.x * 256 + threadIdx.x;
    if (i >= rows * cols) return;
    int r = i / cols, j = i % cols;
    float acc = bias[j];
    const float* x = X + r * 512;
    const float* w = W + j * 512;
    for (int k = 0; k < 512; ++k) acc += x[k] * w[k];
    out[i] = doRelu ? fmaxf(acc, 0.f) : acc;
}

// ---------------------------------------------------------------------------
extern "C" void kernel_launch(void* const* d_in, const int* in_sizes, int n_in,
                              void* d_out, int out_size, void* d_ws, size_t ws_size,
                              hipStream_t stream)
{
    (void)in_sizes; (void)n_in; (void)out_size; (void)ws_size;
    const float* ctx    = (const float*)d_in[0];
    const float* frame  = (const float*)d_in[1];
    const float* nl_vw[2] = {(const float*)d_in[4],  (const float*)d_in[11]};
    const float* nl_ow[2] = {(const float*)d_in[5],  (const float*)d_in[12]};
    const float* nl_ob[2] = {(const float*)d_in[6],  (const float*)d_in[13]};
    const float* nl_g[2]  = {(const float*)d_in[7],  (const float*)d_in[14]};
    const float* nl_be[2] = {(const float*)d_in[8],  (const float*)d_in[15]};
    const float* tp_w  = (const float*)d_in[16];
    const float* tp_b  = (const float*)d_in[17];
    const float* up1_w = (const float*)d_in[18];
    const float* up1_b = (const float*)d_in[19];
    const float* up2_w = (const float*)d_in[20];
    const float* up2_b = (const float*)d_in[21];
    const float* og1_w = (const float*)d_in[22];
    const float* og1_b = (const float*)d_in[23];
    const float* og2_w = (const float*)d_in[24];
    const float* og2_b = (const float*)d_in[25];
    const float* os1_w = (const float*)d_in[26];
    const float* os1_b = (const float*)d_in[27];
    const float* os2_w = (const float*)d_in[28];
    const float* os2_b = (const float*)d_in[29];

    float* ws   = (float*)d_ws;
    float* bufA = ws;                  // 4*1024*1024  x (ping)
    float* bufB = ws + 4194304;        // 4*1024*1024  y (pong)
    float* bufV = ws + 8388608;        // 4*512*1024   V / tp_relu
    float* bufO = ws + 10485760;       // 4*1024*1024  o / s1
    float* bufE = ws + 14680064;       // 24*512*1024  s2
    float* bufPE= ws + 27262976;       // 1024*512     positional encoding
    float* gp   = ws + 27787264;       // 4*512        goal_pre
    float* sga  = ws + 27789312;       // 24*512
    float* gh   = ws + 27801600;       // 4*512
    float* sh   = ws + 27803648;       // 24*512
    float* out  = (float*)d_out;       // goal [4,128] ++ state [24,128]

    // context [B,Tc,C,H,W] -> x [B,C,Tc*H*W]
    k_transpose<<<16384, 256, 0, stream>>>(ctx, bufA);

    // Two non-local blocks. softmax(KQ,axis=2).sum(axis=2) == 1 -> Vo == V,
    // so K/Q/attention drop out exactly; only V and output projections remain.
    float* xin = bufA; float* ybuf = bufB;
    for (int l = 0; l < 2; ++l) {
        k_gemm<<<dim3(32, 16, 4), 32, 0, stream>>>(nl_vw[l], xin, bufV,
                                                   1024, 1024, 1048576, 524288, 1024);
        k_gemm<<<dim3(32, 32, 4), 32, 0, stream>>>(nl_ow[l], bufV, bufO,
                                                   512, 1024, 524288, 1048576, 1024);
        k_addrelu<<<16384, 256, 0, stream>>>(xin, bufO, nl_ob[l], ybuf);
        k_bn<<<1024, 256, 0, stream>>>(ybuf, nl_g[l], nl_be[l]);
        float* tmp = xin; xin = ybuf; ybuf = tmp;
    }

    // Temporal pool conv3d + relu (fused) then spatial mean -> goal_pre
    k_conv3d<<<dim3(13, 16, 4), 32, 0, stream>>>(xin, tp_w, tp_b, bufV);
    k_tpred<<<2048, 256, 0, stream>>>(bufV, gp);

    // Frame branch: two ConvTranspose2d(k2,s2) as GEMMs
    k_convt2<<<dim3(48, 64), 32, 0, stream>>>(up1_w, frame, up1_b, bufO,
                                              1024, 64, 8, 1);
    k_convt2<<<dim3(192, 64), 32, 0, stream>>>(up2_w, bufO, up2_b, bufE,
                                               512, 256, 16, 0);

    // Positional encoding + attention pooling
    k_pe<<<2048, 256, 0, stream>>>(bufPE);
    k_attn<<<24, 256, 0, stream>>>(bufE, gp, bufPE, sga);

    // Heads
    k_head<<<(4 * 512 + 255) / 256, 256, 0, stream>>>(gp,  og1_w, og1_b, gh, 4, 512, 1);
    k_head<<<(4 * 128 + 255) / 256, 256, 0, stream>>>(gh,  og2_w, og2_b, out, 4, 128, 0);
    k_head<<<(24 * 512 + 255) / 256, 256, 0, stream>>>(sga, os1_w, os1_b, sh, 24, 512, 1);
    k_head<<<(24 * 128 + 255) / 256, 256, 0, stream>>>(sh,  os2_w, os2_b, out + 512, 24, 128, 0);
}